// TemporalSummarizer_57535381897478
// MI455X (gfx1250) — compile-verified
//
#include <hip/hip_runtime.h>
#include <math.h>

typedef float v2f __attribute__((ext_vector_type(2)));
typedef float v8f __attribute__((ext_vector_type(8)));

#define EPS 1e-6f

// A: (8,16,64,64,64) f32; elem strides b:4194304, m:262144, t:4096, h:64, w:1
// out: (8,80,64,64) f32; channels: m=A_last, 16+m=A_fast, 32+m=A_slow,
//                                  48+m=A_deriv, 64+m=E_temp
//
// One wave32 per (b,m,h, 16-wide w tile). 16 chained V_WMMA_F32_16X16X4_F32
// accumulate 4 weighted reductions over t (fast EMA, slow EMA, last, deriv)
// in A-matrix rows 0..3, so A streams through HBM exactly once.
// Roofline: ~146 MiB total traffic @ 23.3 TB/s -> ~6.3 us; compute is free.
__global__ __launch_bounds__(256)
void ts_wmma_pass(const float* __restrict__ A, float* __restrict__ out) {
    const int lane = threadIdx.x & 31;
    const int wv   = threadIdx.x >> 5;
    const int tile = blockIdx.x * 8 + wv;          // 32768 tiles total
    const int wt = tile & 3;                       // w tile (16 cols each)
    const int h  = (tile >> 2) & 63;
    const int m  = (tile >> 8) & 15;
    const int b  = tile >> 12;

    const int kh  = lane >> 4;   // lane half: K pair {0,1} (kh=0) or {2,3} (kh=1)
    const int sub = lane & 15;   // A-matrix row M / B-matrix col N

    // EMA weights: w(t) = (1-alpha)*alpha^(63-t); K-step advances t by 4.
    const float e0  = (float)(63 - 2 * kh);
    const float wf0 = 0.26f * __powf(0.74f, e0);
    const float ws0 = 0.05f * __powf(0.95f, e0);
    const float stepF = 1.0f / (0.74f * 0.74f * 0.74f * 0.74f);
    const float stepS = 1.0f / (0.95f * 0.95f * 0.95f * 0.95f);

    // Branchless per-lane A-matrix row state (rows 2..15 stay zero in-loop;
    // 0 * st == 0 so the step factor for those rows is a don't-care).
    float       ax   = (sub == 0) ? wf0 : ((sub == 1) ? ws0 : 0.0f);
    const float invA = (sub == 1) ? (1.0f / 0.95f) : (1.0f / 0.74f);
    float       ay   = ax * invA;              // K+1 weight = K weight / alpha
    const float st   = (sub == 1) ? stepS : stepF;

    // Final-step additive selectors (only lane-half 1 sees t=62,63):
    //   row 2: delta(t=63) -> A_last ; row 3: delta(63)-delta(62) -> A_deriv
    const bool  hi   = (kh == 1);
    const float addx = (hi && sub == 3) ? -1.0f : 0.0f;
    const float addy = (hi && (sub == 2 || sub == 3)) ? 1.0f : 0.0f;

    // &A[b,m, t=2*kh, h, wt*16 + sub]
    const float* base = A + (size_t)(b * 16 + m) * 262144
                          + (size_t)h * 64 + (size_t)(wt * 16 + sub)
                          + (size_t)(2 * kh) * 4096;

    v8f acc = {};
    #pragma unroll
    for (int k = 0; k < 15; ++k) {
        // B-matrix f32 4x16 layout: VGPR0 = K rows {0,2}, VGPR1 = {1,3}.
        // Compiler hoists all these loads into one up-front clause, giving
        // ~30 in-flight NT loads per wave; no explicit prefetch needed.
        const float* p = base + (size_t)(4 * k) * 4096;
        v2f bv;
        bv.x = __builtin_nontemporal_load(p);          // K = 4k + 2*kh
        bv.y = __builtin_nontemporal_load(p + 4096);   // K = 4k + 2*kh + 1

        v2f av; av.x = ax; av.y = ay;
        acc = __builtin_amdgcn_wmma_f32_16x16x4_f32(
            false, av, false, bv, (short)0, acc, false, false);

        ax *= st; ay *= st;
    }
    {   // peeled final K-step (t = 60..63): add last/deriv selector rows
        const float* p = base + (size_t)60 * 4096;
        v2f bv;
        bv.x = __builtin_nontemporal_load(p);
        bv.y = __builtin_nontemporal_load(p + 4096);
        v2f av; av.x = ax + addx; av.y = ay + addy;
        acc = __builtin_amdgcn_wmma_f32_16x16x4_f32(
            false, av, false, bv, (short)0, acc, false, false);
    }

    // D layout: acc[r], lanes 0-15 = row M=r, N=lane. Rows 0..3 live.
    if (lane < 16) {
        const size_t spatial = (size_t)h * 64 + (size_t)(wt * 16 + lane);
        float* ob = out + (size_t)b * 80 * 4096 + spatial;
        __builtin_nontemporal_store(acc[2], ob + (size_t)(m)      * 4096); // last
        __builtin_nontemporal_store(acc[0], ob + (size_t)(16 + m) * 4096); // fast
        __builtin_nontemporal_store(acc[1], ob + (size_t)(32 + m) * 4096); // slow
        __builtin_nontemporal_store(acc[3], ob + (size_t)(48 + m) * 4096); // deriv
    }
}

// E_temp from U[:, :, comp, t in {61,63}, h, w] -- elementwise, ~8 MiB of reads.
__global__ __launch_bounds__(256)
void ts_energy_pass(const float* __restrict__ U, float* __restrict__ out) {
    const int idx = blockIdx.x * 256 + threadIdx.x;   // 524288 = 8*16*4096
    const int s  = idx & 4095;        // h*64 + w
    const int bm = idx >> 12;         // b*16 + m
    const int m  = bm & 15;
    const int b  = bm >> 4;

    const size_t comp = 262144;       // 64*64*64
    const float* u0 = U + (size_t)bm * 2 * comp;      // real
    const float* u1 = u0 + comp;                      // imag

    const float r1 = __builtin_nontemporal_load(u0 + (size_t)63 * 4096 + s);
    const float r3 = __builtin_nontemporal_load(u0 + (size_t)61 * 4096 + s);
    const float i1 = __builtin_nontemporal_load(u1 + (size_t)63 * 4096 + s);
    const float i3 = __builtin_nontemporal_load(u1 + (size_t)61 * 4096 + s);

    const float qr = 0.5f * (r1 - r3);
    const float qi = 0.5f * (i1 - i3);
    const float er2 = fmaxf(r1 * r1 + qr * qr, EPS);  // E_re^2
    const float ei2 = fmaxf(i1 * i1 + qi * qi, EPS);  // E_im^2
    const float e = sqrtf(fmaxf(er2 + ei2, EPS));

    __builtin_nontemporal_store(
        e, out + ((size_t)(b * 80 + 64 + m)) * 4096 + (size_t)s);
}

extern "C" void kernel_launch(void* const* d_in, const int* in_sizes, int n_in,
                              void* d_out, int out_size, void* d_ws, size_t ws_size,
                              hipStream_t stream) {
    (void)in_sizes; (void)n_in; (void)out_size; (void)d_ws; (void)ws_size;
    const float* A = (const float*)d_in[0];
    const float* U = (const float*)d_in[1];
    float* out = (float*)d_out;

    // 8*16*64*4 = 32768 wave-tiles, 8 waves per 256-thread block.
    ts_wmma_pass<<<4096, 256, 0, stream>>>(A, out);
    // 8*16*64*64 = 524288 elements.
    ts_energy_pass<<<2048, 256, 0, stream>>>(U, out);
}